// VB_67216238182491
// MI455X (gfx1250) — compile-verified
//
#include <hip/hip_runtime.h>

constexpr int NN = 4096;   // fixed by the reference (N); compile-time so row
                           // strides fold into VMEM instruction offsets
#define KTAB 501
#define NB 64
#define LDT 66   // padded LDS stride for SYRK tiles (even -> 8B-aligned float2 lds reads)

typedef float v2f __attribute__((ext_vector_type(2)));
typedef float v8f __attribute__((ext_vector_type(8)));

__device__ __forceinline__ int iclampi(int v, int lo, int hi) {
    return v < lo ? lo : (v > hi ? hi : v);
}

// Bilinear interp on (uniform mu grid) x (log-uniform s2 grid), grids in LDS,
// 1MB fs table via L2-cached global gathers.
__device__ __forceinline__ float vb_interp2(const float* __restrict__ fs,
                                            const float* __restrict__ xg,
                                            const float* __restrict__ yg,
                                            float x, float y) {
    const float X0 = -10.0f;                 // MU_LIMS = (-5*SIGMA0, 5*SIGMA0)
    const float INV_DX = 500.0f / 20.0f;
    x = fminf(fmaxf(x, X0), 10.0f);
    int i = iclampi((int)floorf((x - X0) * INV_DX), 0, KTAB - 2);
    float tx = (x - xg[i]) / (xg[i + 1] - xg[i]);

    y = fminf(fmaxf(y, yg[0]), yg[KTAB - 1]);
    const float E0 = -4.0f;                  // S2_LIMS in log10
    const float INV_DE = 500.0f / 6.0f;
    int j = iclampi((int)floorf((__log10f(y) - E0) * INV_DE), 0, KTAB - 2);
    if (y < yg[j])           j = iclampi(j - 1, 0, KTAB - 2);
    else if (y >= yg[j + 1]) j = iclampi(j + 1, 0, KTAB - 2);
    float ty = (y - yg[j]) / (yg[j + 1] - yg[j]);

    float f00 = fs[i * KTAB + j];
    float f10 = fs[(i + 1) * KTAB + j];
    float f01 = fs[i * KTAB + j + 1];
    float f11 = fs[(i + 1) * KTAB + j + 1];
    float a0 = f00 + tx * (f10 - f00);
    float a1 = f01 + tx * (f11 - f01);
    return a0 + ty * (a1 - a0);
}

// ---------------------------------------------------------------------------
// Kernel 0: zero the logdet accumulator (ws is poisoned; must init every call)
// ---------------------------------------------------------------------------
__global__ void vb_init_kernel(float* logacc) {
    if (threadIdx.x == 0) logacc[0] = 0.0f;
}

// ---------------------------------------------------------------------------
// Kernel 1: ell = sum_jk [ w*fw + (n-w)*fl ]  (128x128 tile per block)
// Memory-bound: streams n, w, Sigma once (192MB @ 23.3TB/s), fs from L2.
// ---------------------------------------------------------------------------
__global__ void vb_ell_kernel(const float* __restrict__ nmat,
                              const float* __restrict__ wmat,
                              const float* __restrict__ mu,
                              const float* __restrict__ Sigma,
                              const float* __restrict__ mu_grid,
                              const float* __restrict__ s2_grid,
                              const float* __restrict__ fs,
                              float* __restrict__ partials) {
    __shared__ float sxg[KTAB];
    __shared__ float syg[KTAB];
    __shared__ float dr[128], dc[128], mur[128], muc[128];
    __shared__ float red[256];

    const int tid = threadIdx.x;
    for (int i = tid; i < KTAB; i += 256) { sxg[i] = mu_grid[i]; syg[i] = s2_grid[i]; }
    const int j0 = blockIdx.y * 128;
    const int k0 = blockIdx.x * 128;
    for (int i = tid; i < 128; i += 256) {
        int j = j0 + i, k = k0 + i;
        dr[i]  = Sigma[(size_t)j * NN + j];
        dc[i]  = Sigma[(size_t)k * NN + k];
        mur[i] = mu[j];
        muc[i] = mu[k];
    }
    __syncthreads();

    float acc = 0.0f;
    for (int e = tid; e < 128 * 128; e += 256) {
        int r = e >> 7, c = e & 127;
        int j = j0 + r, k = k0 + c;
        size_t idx = (size_t)j * NN + k;
        float nv = nmat[idx];
        float wv = wmat[idx];
        if (j == k) {
            acc += 0.5f * nv;                       // w*0.5 + (n-w)*0.5
        } else {
            float sjk = Sigma[idx];                 // Sigma symmetric: Sjk == Skj
            float md  = mur[r] - muc[c];
            float s2d = dr[r] + dc[c] - 2.0f * sjk;
            float fw = vb_interp2(fs, sxg, syg,  md, s2d);
            float fl = vb_interp2(fs, sxg, syg, -md, s2d);
            acc += wv * fw + (nv - wv) * fl;
        }
    }
    red[tid] = acc;
    __syncthreads();
    for (int s = 128; s > 0; s >>= 1) {
        if (tid < s) red[tid] += red[tid + s];
        __syncthreads();
    }
    if (tid == 0) partials[blockIdx.y * gridDim.x + blockIdx.x] = red[0];
}

// ---------------------------------------------------------------------------
// Kernel 2: unblocked Cholesky of the NBxNB diagonal block (single workgroup),
// accumulates sum(log L_ii) deterministically (one block, stream-ordered).
// ---------------------------------------------------------------------------
__global__ void vb_potf2_kernel(float* __restrict__ M, int kb,
                                float* __restrict__ logacc) {
    __shared__ float Lb[NB * NB];
    const int tid = threadIdx.x;   // blockDim = 64
    for (int i = tid; i < NB * NB; i += 64) {
        int r = i >> 6, c = i & 63;
        Lb[i] = M[(size_t)(kb + r) * NN + (kb + c)];
    }
    __syncthreads();
    float lsum = 0.0f;
    for (int c = 0; c < NB; ++c) {
        if (tid == 0) {
            float s = Lb[c * NB + c];
            for (int t = 0; t < c; ++t) { float v = Lb[c * NB + t]; s -= v * v; }
            float l = sqrtf(fmaxf(s, 1e-30f));
            Lb[c * NB + c] = l;
            lsum += __logf(l);
        }
        __syncthreads();
        int r = c + 1 + tid;
        if (r < NB) {
            float s = Lb[r * NB + c];
            for (int t = 0; t < c; ++t) s -= Lb[r * NB + t] * Lb[c * NB + t];
            Lb[r * NB + c] = s / Lb[c * NB + c];
        }
        __syncthreads();
    }
    for (int i = tid; i < NB * NB; i += 64) {
        int r = i >> 6, c = i & 63;
        M[(size_t)(kb + r) * NN + (kb + c)] = Lb[i];
    }
    if (tid == 0) logacc[0] += lsum;   // sequential across stream-ordered launches
}

// ---------------------------------------------------------------------------
// Kernel 3: panel TRSM: L21 = A21 * L11^{-T}; one row per thread, L11 in LDS,
// per-thread solution vector kept in LDS (conflict-free: stride 65).
// ---------------------------------------------------------------------------
__global__ void vb_trsm_kernel(float* __restrict__ M, int kb) {
    __shared__ float L11[NB * NB];
    __shared__ float xs[64 * 65];
    const int tid = threadIdx.x;   // blockDim = 64
    for (int i = tid; i < NB * NB; i += 64) {
        int r = i >> 6, c = i & 63;
        L11[i] = M[(size_t)(kb + r) * NN + (kb + c)];
    }
    __syncthreads();
    const int gi = kb + NB + blockIdx.x * 64 + tid;
    if (gi < NN) {
        float* x = &xs[tid * 65];
        float* row = &M[(size_t)gi * NN + kb];
        for (int c = 0; c < NB; ++c) {
            float s = row[c];
            for (int t = 0; t < c; ++t) s -= x[t] * L11[c * NB + t];
            x[c] = s / L11[c * NB + c];
        }
        for (int c = 0; c < NB; ++c) row[c] = x[c];
    }
}

// ---------------------------------------------------------------------------
// Kernel 4: WMMA SYRK trailing update: A22 -= L21 * L21^T (lower tiles only).
// 64x64 C tile / block of 128 threads (4 waves). Each wave register-blocks a
// 32x32 quadrant: 4 accumulators, 2 A-frags + 2 B-frags -> 4 wmma per k-step
// (1:1 ds:wmma). Writeback: batched loads, single wait, batched stores;
// compile-time N folds row strides into VMEM immediate offsets.
// ---------------------------------------------------------------------------
__global__ void vb_syrk_wmma_kernel(float* __restrict__ M, int kb) {
    const int ti = blockIdx.x;
    const int tj = blockIdx.y;
    if (tj > ti) return;                       // uniform block exit: EXEC all-ones
    const int base  = kb + NB;
    const int arow0 = base + ti * 64;
    const int brow0 = base + tj * 64;

    __shared__ float As[64 * LDT];
    __shared__ float Bs[64 * LDT];
    const int tid = threadIdx.x;               // blockDim = 128 (4 waves)
    for (int i = tid; i < 64 * 64; i += 128) {
        int r = i >> 6, c = i & 63;
        As[r * LDT + c] = M[(size_t)(arow0 + r) * NN + (kb + c)];
        Bs[r * LDT + c] = M[(size_t)(brow0 + r) * NN + (kb + c)];
    }
    __syncthreads();

    const int wave = tid >> 5;
    const int lane = tid & 31;
    const int half = lane >> 4;                // wave32: lanes 0-15 -> K{0,1}, 16-31 -> K{2,3}
    const int m    = lane & 15;
    const int rq   = (wave >> 1) * 32;         // 32x32 quadrant of the 64x64 tile
    const int cq   = (wave & 1) * 32;

    v8f acc00 = {}, acc01 = {}, acc10 = {}, acc11 = {};
    const int kofs = 2 * half;
    #pragma unroll
    for (int k = 0; k < NB; k += 4) {
        const float* a0p = &As[(rq + m) * LDT + k + kofs];
        const float* a1p = &As[(rq + 16 + m) * LDT + k + kofs];
        const float* b0p = &Bs[(cq + m) * LDT + k + kofs];
        const float* b1p = &Bs[(cq + 16 + m) * LDT + k + kofs];
        v2f a0; a0[0] = a0p[0]; a0[1] = a0p[1];
        v2f a1; a1[0] = a1p[0]; a1[1] = a1p[1];
        v2f b0; b0[0] = b0p[0]; b0[1] = b0p[1];
        v2f b1; b1[0] = b1p[0]; b1[1] = b1p[1];
        acc00 = __builtin_amdgcn_wmma_f32_16x16x4_f32(false, a0, false, b0, (short)0, acc00, false, false);
        acc01 = __builtin_amdgcn_wmma_f32_16x16x4_f32(false, a0, false, b1, (short)0, acc01, false, false);
        acc10 = __builtin_amdgcn_wmma_f32_16x16x4_f32(false, a1, false, b0, (short)0, acc10, false, false);
        acc11 = __builtin_amdgcn_wmma_f32_16x16x4_f32(false, a1, false, b1, (short)0, acc11, false, false);
    }

    // C/D layout: VGPR r -> row (sub + r + 8*half), col (sub + m).
    // Batched RMW: 8 loads (pipelined), subtract, 8 stores; strides are
    // compile-time (r*NN*4 bytes) -> single base address + immediate offsets.
    #pragma unroll
    for (int qr = 0; qr < 2; ++qr) {
        #pragma unroll
        for (int qc = 0; qc < 2; ++qc) {
            const v8f& acc = qr == 0 ? (qc == 0 ? acc00 : acc01)
                                     : (qc == 0 ? acc10 : acc11);
            const int gr0 = arow0 + rq + qr * 16 + 8 * half;
            const int gc  = brow0 + cq + qc * 16 + m;
            float* p = &M[(size_t)gr0 * NN + gc];
            float cv[8];
            #pragma unroll
            for (int r = 0; r < 8; ++r) cv[r] = p[(size_t)r * NN];
            #pragma unroll
            for (int r = 0; r < 8; ++r) p[(size_t)r * NN] = cv[r] - acc[r];
        }
    }
}

// ---------------------------------------------------------------------------
// Kernel 5: combine: sum(ell partials) + prior + entropy -> d_out[0]
// ---------------------------------------------------------------------------
__global__ void vb_final_kernel(const float* __restrict__ partials, int nparts,
                                const float* __restrict__ mu,
                                const float* __restrict__ Sigma,
                                const float* __restrict__ logacc,
                                float* __restrict__ out) {
    __shared__ float red[256];
    const int tid = threadIdx.x;
    float a = 0.0f;
    for (int i = tid; i < nparts; i += 256) a += partials[i];
    float p = 0.0f;
    for (int i = tid; i < NN; i += 256) {
        float m = mu[i];                        // MU0 == 0
        p += m * m + Sigma[(size_t)i * NN + i];
    }
    a += -0.25f * p;                            // -1/(2*SIGMA0), SIGMA0 = 2
    red[tid] = a;
    __syncthreads();
    for (int s = 128; s > 0; s >>= 1) {
        if (tid < s) red[tid] += red[tid + s];
        __syncthreads();
    }
    if (tid == 0) {
        const float LOG2PIE = 2.8378770664093453f;       // log(2*pi*e)
        // entropy = 0.5*(N*log(2pi e) + logdet); logdet = 2*sum(log L_ii)
        out[0] = red[0] + 0.5f * (float)NN * LOG2PIE + logacc[0];
    }
}

// ---------------------------------------------------------------------------
extern "C" void kernel_launch(void* const* d_in, const int* in_sizes, int n_in,
                              void* d_out, int out_size, void* d_ws, size_t ws_size,
                              hipStream_t stream) {
    const float* nmat    = (const float*)d_in[0];   // (N,N)
    const float* wmat    = (const float*)d_in[1];   // (N,N)
    const float* mu      = (const float*)d_in[2];   // (N,)
    const float* Sigma   = (const float*)d_in[3];   // (N,N)
    const float* mu_grid = (const float*)d_in[4];   // (K,)
    const float* s2_grid = (const float*)d_in[5];   // (K,)
    const float* fs      = (const float*)d_in[6];   // (K,K)
    // d_in[7..10] (derivative tables) are unused by the objective value.

    float* out = (float*)d_out;

    // workspace: [ Mchol (N*N) | partials (tiles^2) | logacc (1) ]
    float* Mchol = (float*)d_ws;
    const int tiles = NN / 128;
    float* partials = Mchol + (size_t)NN * NN;
    float* logacc   = partials + (size_t)tiles * tiles;

    vb_init_kernel<<<1, 64, 0, stream>>>(logacc);

    // In-place Cholesky needs a mutable copy (must not mutate d_in).
    hipMemcpyAsync(Mchol, Sigma, (size_t)NN * NN * sizeof(float),
                   hipMemcpyDeviceToDevice, stream);

    // ell: memory-bound streaming pass over n, w, Sigma with table gathers.
    dim3 egrid(tiles, tiles);
    vb_ell_kernel<<<egrid, 256, 0, stream>>>(nmat, wmat, mu, Sigma,
                                             mu_grid, s2_grid, fs, partials);

    // Blocked Cholesky for logdet; SYRK trailing update uses FP32 WMMA.
    for (int kb = 0; kb < NN; kb += NB) {
        vb_potf2_kernel<<<1, 64, 0, stream>>>(Mchol, kb, logacc);
        const int rem = NN - kb - NB;
        if (rem > 0) {
            vb_trsm_kernel<<<rem / 64, 64, 0, stream>>>(Mchol, kb);
            const int mt = rem / 64;
            vb_syrk_wmma_kernel<<<dim3(mt, mt), 128, 0, stream>>>(Mchol, kb);
        }
    }

    vb_final_kernel<<<1, 256, 0, stream>>>(partials, tiles * tiles, mu, Sigma,
                                           logacc, out);
}